// SimpleCRF_35150012350673
// MI455X (gfx1250) — compile-verified
//
#include <hip/hip_runtime.h>
#include <stdint.h>

// Viterbi CRF decode for MI455X (gfx1250).  B=512, S=512, T=128.
// Max-plus semiring -> WMMA not applicable; kernel is VALU bound by design.
// CDNA5 features used:
//   * global_load_async_to_lds_b128 + s_wait_asynccnt: async DMA of the 64KB
//     transition matrix into LDS, overlapped with score init.
//   * Large register file (up to 1024 VGPRs/wave): each lane caches its whole
//     128-float transition COLUMN in VGPRs, eliminating 128 ds_loads per
//     timestep (65k ds_loads per lane over the sequence).
//   * 320KB LDS/WGP: 64KB trans staging buffer UNIONed with the 64KB
//     backpointer history -> backtrace runs on LDS latency, and the block
//     still fits 4-per-WGP.
//   * wave32: 128-thread block = 4 waves.

#define NUM_TAGS 128
#define BATCH    512
#define SEQ      512

// Phase 1: async-DMA staging of transitions.  Phase 2 (after columns are
// extracted into VGPRs): backpointer history.  Never live simultaneously.
union SharedOverlay {
    float   trans[NUM_TAGS * NUM_TAGS];     // 64 KB
    uint8_t hist[(SEQ - 1) * NUM_TAGS];     // 511*128 = 63.9 KB
};

__global__ __launch_bounds__(NUM_TAGS)
void viterbi_crf_gfx1250(const float*   __restrict__ emis,     // [B,S,T]
                         const uint8_t* __restrict__ mask,     // [B,S] (jax bool, 1B)
                         const float*   __restrict__ start_t,  // [T]
                         const float*   __restrict__ end_t,    // [T]
                         const float*   __restrict__ trans,    // [T,T] row-major
                         int*           __restrict__ out_paths,// [B,S]
                         float*         __restrict__ out_score)// [B]
{
    __shared__ __align__(16) SharedOverlay u;                 // 64 KB
    __shared__ __align__(16) float         s_score[2][NUM_TAGS];
    __shared__ int                         s_path[SEQ];
    __shared__ float                       s_final[NUM_TAGS];
    __shared__ int                         s_red[NUM_TAGS];

    const int    j     = threadIdx.x;            // next-tag owned by this lane
    const int    b     = blockIdx.x;             // batch element for this block
    const size_t ebase = (size_t)b * SEQ * NUM_TAGS;
    const size_t mbase = (size_t)b * SEQ;

    // ---- Phase 1: async-DMA transitions (64 KB) into LDS ----
    // 4096 16-byte chunks; 128 lanes x 32 iterations, one b128 async copy each.
    #pragma unroll 4
    for (int k = 0; k < 32; ++k) {
        const int    chunk = k * NUM_TAGS + j;
        unsigned     ldsa  = (unsigned)(uintptr_t)(&u.trans[chunk * 4]);
        const float* gsrc  = trans + (size_t)chunk * 4;
        asm volatile("global_load_async_to_lds_b128 %0, %1, off"
                     :: "v"(ldsa), "v"(gsrc) : "memory");
    }

    // Overlap DMA with score init: score0[j] = start[j] + e[b,0,j]
    float own = start_t[j] + emis[ebase + j];
    s_score[0][j] = own;

    asm volatile("s_wait_asynccnt 0" ::: "memory");   // DMA landed (this wave)
    __syncthreads();                                  // visible to all 4 waves

    // ---- Extract this lane's transition column into VGPRs ----
    // tc[i] = trans[i][j].  dword address i*128+j -> bank j%64: conflict-free.
    float tc[NUM_TAGS];
    #pragma unroll
    for (int i = 0; i < NUM_TAGS; ++i)
        tc[i] = u.trans[i * NUM_TAGS + j];

    __syncthreads();   // trans staging dead; LDS region becomes hist

    // ---- Forward max-plus recurrence (VALU-bound, 34 DS ops/step) ----
    int     cur    = 0;
    float   e_next = emis[ebase + (size_t)NUM_TAGS + j];   // e[b,1,j]
    uint8_t m_next = mask[mbase + 1];

    for (int t = 1; t < SEQ; ++t) {
        const float   e_t = e_next;
        const uint8_t m_t = m_next;
        if (t + 1 < SEQ) {   // register prefetch of next emission row
            e_next = emis[ebase + (size_t)(t + 1) * NUM_TAGS + j];
            m_next = mask[mbase + t + 1];
        }

        float best = -3.402823466e38f;
        int   bi   = 0;
        const float4* sc4 = reinterpret_cast<const float4*>(s_score[cur]);
        #pragma unroll
        for (int q = 0; q < NUM_TAGS / 4; ++q) {
            const float4 sv = sc4[q];          // LDS broadcast (same addr all lanes)
            const int    i0 = q * 4;
            float c0 = sv.x + tc[i0 + 0];
            float c1 = sv.y + tc[i0 + 1];
            float c2 = sv.z + tc[i0 + 2];
            float c3 = sv.w + tc[i0 + 3];
            // strictly-greater keeps FIRST max (matches jnp.argmax)
            if (c0 > best) { best = c0; bi = i0;     }
            if (c1 > best) { best = c1; bi = i0 + 1; }
            if (c2 > best) { best = c2; bi = i0 + 2; }
            if (c3 > best) { best = c3; bi = i0 + 3; }
        }

        u.hist[(t - 1) * NUM_TAGS + j] = (uint8_t)bi;   // backpointer
        own = m_t ? (best + e_t) : own;                 // masked update
        s_score[cur ^ 1][j] = own;
        __syncthreads();
        cur ^= 1;
    }

    // ---- Final scores + seq_end reduction ----
    s_final[j] = own + end_t[j];
    int msum = 0;
    #pragma unroll
    for (int k = j; k < SEQ; k += NUM_TAGS) msum += (int)mask[mbase + k];
    s_red[j] = msum;
    __syncthreads();

    // ---- Backtrace entirely in LDS ----
    if (j == 0) {
        int seq_end = -1;
        for (int k = 0; k < NUM_TAGS; ++k) seq_end += s_red[k];

        float bs = s_final[0];
        int   bl = 0;
        for (int k = 1; k < NUM_TAGS; ++k)
            if (s_final[k] > bs) { bs = s_final[k]; bl = k; }
        out_score[b] = bs;

        int tag = bl;
        for (int t = SEQ - 1; t >= 1; --t) {
            s_path[t] = tag;
            const int prev = (int)u.hist[(t - 1) * NUM_TAGS + tag];
            tag = (t <= seq_end) ? prev : tag;
        }
        s_path[0] = tag;
    }
    __syncthreads();

    // Coalesced path writeback, masked positions -> -1 (reference semantics)
    #pragma unroll
    for (int k = j; k < SEQ; k += NUM_TAGS) {
        const int p = s_path[k];
        out_paths[mbase + k] = mask[mbase + k] ? p : -1;
    }
}

extern "C" void kernel_launch(void* const* d_in, const int* in_sizes, int n_in,
                              void* d_out, int out_size, void* d_ws, size_t ws_size,
                              hipStream_t stream) {
    (void)in_sizes; (void)n_in; (void)d_ws; (void)ws_size; (void)out_size;

    const float*   emis    = (const float*)d_in[0];    // [B,S,T] f32
    const uint8_t* mask    = (const uint8_t*)d_in[1];  // [B,S] bool (1 byte)
    const float*   start_t = (const float*)d_in[2];    // [T]
    const float*   end_t   = (const float*)d_in[3];    // [T]
    const float*   trans   = (const float*)d_in[4];    // [T,T]

    // d_out = paths [B,S] int32 (raw bits) followed by best_score [B] f32
    int*   out_paths = (int*)d_out;
    float* out_score = (float*)d_out + (size_t)BATCH * SEQ;

    viterbi_crf_gfx1250<<<BATCH, NUM_TAGS, 0, stream>>>(
        emis, mask, start_t, end_t, trans, out_paths, out_score);
}